// DiceLoss_21749714387527
// MI455X (gfx1250) — compile-verified
//
#include <hip/hip_runtime.h>

// DiceLoss: B=8, C=21, H=W=512, inputs f32, targets int64, smooth int scalar.
// Stage 1: streaming softmax + masked partial sums (BW-bound, ~193MB @ 23.3TB/s ~ 8.3us floor).
// Stage 2: single-wave finalize; block-partial reduction done with v_wmma_f32_16x16x4_f32
//          (A = ones  =>  D[m][n] = sum_k B[k][n] + C[m][n], layout-invariant total).

#define NCH 21
#define IGN 255
#define PPT 4            // pixels per thread
#define TPB 256          // threads per block (8 wave32 waves)

typedef __attribute__((ext_vector_type(2))) float v2f;
typedef __attribute__((ext_vector_type(8))) float v8f;

__global__ __launch_bounds__(TPB) void dice_partial_kernel(
    const float* __restrict__ x, const long long* __restrict__ tgt,
    float* __restrict__ part, int HW, int blocksPerBatch, int nBlocks)
{
  const int b    = blockIdx.x / blocksPerBatch;
  const int blk  = blockIdx.x - b * blocksPerBatch;
  const int base = blk * (TPB * PPT) + threadIdx.x * PPT;

  const float*     __restrict__ xb = x   + (size_t)b * NCH * HW;
  const long long* __restrict__ tb = tgt + (size_t)b * HW;

  // All 21 channels of 4 pixels live in registers: one pass over memory.
  float v[NCH][PPT];
  #pragma unroll
  for (int c = 0; c < NCH; ++c) {
    float4 ld = *reinterpret_cast<const float4*>(xb + (size_t)c * HW + base);
    v[c][0] = ld.x; v[c][1] = ld.y; v[c][2] = ld.z; v[c][3] = ld.w;
  }

  float inter = 0.f, psum = 0.f, cnt = 0.f;
  #pragma unroll
  for (int j = 0; j < PPT; ++j) {
    const int   tj = (int)tb[base + j];
    const float mf = (tj != IGN) ? 1.0f : 0.0f;

    float m = v[0][j];
    #pragma unroll
    for (int c = 1; c < NCH; ++c) m = fmaxf(m, v[c][j]);

    float s = 0.f, et = 0.f;
    #pragma unroll
    for (int c = 0; c < NCH; ++c) {
      const float e = __expf(v[c][j] - m);
      s += e;
      et = (c == tj) ? e : et;      // gather target prob numerator by select
    }
    const float rcp = 1.0f / s;
    inter += et * rcp * mf;          // softmax prob of target class
    psum  += s  * rcp * mf;          // sum_c softmax (== 1 up to rounding)
    cnt   += mf;
  }

  // wave32 tree reduction
  #pragma unroll
  for (int off = 16; off > 0; off >>= 1) {
    inter += __shfl_xor(inter, off, 32);
    psum  += __shfl_xor(psum,  off, 32);
    cnt   += __shfl_xor(cnt,   off, 32);
  }
  __shared__ float red[3][TPB / 32];
  const int wave = threadIdx.x >> 5;
  if ((threadIdx.x & 31) == 0) {
    red[0][wave] = inter; red[1][wave] = psum; red[2][wave] = cnt;
  }
  __syncthreads();
  if (threadIdx.x == 0) {
    float a = 0.f, p2 = 0.f, c2 = 0.f;
    #pragma unroll
    for (int w = 0; w < TPB / 32; ++w) { a += red[0][w]; p2 += red[1][w]; c2 += red[2][w]; }
    // layout: part[q * nBlocks + b * blocksPerBatch + blk] -> 256 contiguous per (q,b)
    part[0 * nBlocks + blockIdx.x] = a;
    part[1 * nBlocks + blockIdx.x] = p2;
    part[2 * nBlocks + blockIdx.x] = c2;
  }
}

__global__ __launch_bounds__(32) void dice_final_kernel(
    const float* __restrict__ part, const int* __restrict__ smooth,
    float* __restrict__ out, int blocksPerBatch, int nBlocks)
{
  const int lane = threadIdx.x;
  v2f ones; ones[0] = 1.0f; ones[1] = 1.0f;
  float sums[3][8];
  const int chunks = blocksPerBatch >> 6;   // 64 values per WMMA step

  // EXEC is all-1s here (32 uniform threads, no divergence before the WMMAs).
  for (int q = 0; q < 3; ++q) {
    for (int b = 0; b < 8; ++b) {
      const float* p = part + q * nBlocks + b * blocksPerBatch;
      v8f acc = {};
      for (int ch = 0; ch < chunks; ++ch) {
        v2f bm;
        bm[0] = p[ch * 64 + lane];
        bm[1] = p[ch * 64 + 32 + lane];
        // A = ones => D[m][n] = sum_k B[k][n] + C[m][n]; total is layout-invariant.
        acc = __builtin_amdgcn_wmma_f32_16x16x4_f32(
            /*neg_a=*/false, ones, /*neg_b=*/false, bm,
            /*c_mod=*/(short)0, acc, /*reuse_a=*/false, /*reuse_b=*/false);
      }
      // D VGPR0 holds rows {0,8}: each column sum appears twice across 32 lanes.
      float r = acc[0];
      #pragma unroll
      for (int off = 16; off > 0; off >>= 1) r += __shfl_xor(r, off, 32);
      sums[q][b] = 0.5f * r;
    }
  }

  if (lane == 0) {
    const float sm = (float)smooth[0];
    float acc = 0.f;
    #pragma unroll
    for (int b = 0; b < 8; ++b)
      acc += 1.0f - (2.0f * sums[0][b] + sm) / (sums[1][b] + sums[2][b] + sm);
    out[0] = acc * (1.0f / 8.0f);
  }
}

extern "C" void kernel_launch(void* const* d_in, const int* in_sizes, int n_in,
                              void* d_out, int out_size, void* d_ws, size_t ws_size,
                              hipStream_t stream) {
  const float*     x      = (const float*)d_in[0];
  const long long* t      = (const long long*)d_in[1];
  const int*       smooth = (const int*)d_in[2];
  float* out  = (float*)d_out;
  float* part = (float*)d_ws;            // 3 * nBlocks floats = 24 KB

  const int B  = 8;
  const int P  = in_sizes[1];            // B*H*W = 2,097,152
  const int HW = P / B;                  // 262,144
  const int blocksPerBatch = HW / (TPB * PPT);   // 256
  const int nBlocks        = B * blocksPerBatch; // 2048

  dice_partial_kernel<<<nBlocks, TPB, 0, stream>>>(x, t, part, HW, blocksPerBatch, nBlocks);
  dice_final_kernel<<<1, 32, 0, stream>>>(part, smooth, out, blocksPerBatch, nBlocks);
}